// FraudDetectionGNN_86973087744532
// MI455X (gfx1250) — compile-verified
//
#include <hip/hip_runtime.h>
#include <cstdint>
#include <cstddef>

#define NEG_SLOPE 0.2f

typedef float v2f __attribute__((ext_vector_type(2)));
typedef float v8f __attribute__((ext_vector_type(8)));

static inline int cdiv(long long a, long long b) { return (int)((a + b - 1) / b); }

// ------------------------------------------------------------------
// fp32 GEMM on the WMMA pipe: C[M,Nd] = A[M,Kd] @ B[Kd,Nd]
// One wave computes one 16x16 output tile via V_WMMA_F32_16X16X4_F32.
// A frag (16x4, 2 VGPR): lane&15 -> M row; half-wave g -> K base {0,2}; vgpr -> +1
// B frag (4x16, 2 VGPR): lane&15 -> N col; same K pattern
// C/D  (16x16, 8 VGPR): lane&15 -> N col; row = 8*g + vgpr
// Requires M % 16 == 0 (true here: 50000 = 3125*16).
// ------------------------------------------------------------------
template <int Nd, int Kd>
__global__ void gemm_f32_wmma(const float* __restrict__ A,
                              const float* __restrict__ B,
                              float* __restrict__ C, int M) {
  const int warp = (blockIdx.x * blockDim.x + threadIdx.x) >> 5;
  const int lane = threadIdx.x & 31;
  constexpr int NT = Nd / 16;
  const int tm = warp / NT, tn = warp % NT;
  if (tm * 16 >= M) return;  // uniform across the wave -> EXEC stays all-1s
  const int col = lane & 15;
  const int g = lane >> 4;

  const float* Arow = A + (size_t)(tm * 16 + col) * Kd;  // lane&15 = A row (M)
  const float* Bcol = B + tn * 16 + col;                 // lane&15 = B col (N)

  v8f acc = {0.f, 0.f, 0.f, 0.f, 0.f, 0.f, 0.f, 0.f};
#pragma unroll 4
  for (int k0 = 0; k0 < Kd; k0 += 4) {
    v2f a, b;
    a.x = Arow[k0 + 2 * g + 0];
    a.y = Arow[k0 + 2 * g + 1];
    b.x = Bcol[(size_t)(k0 + 2 * g + 0) * Nd];
    b.y = Bcol[(size_t)(k0 + 2 * g + 1) * Nd];
    acc = __builtin_amdgcn_wmma_f32_16x16x4_f32(false, a, false, b,
                                                (short)0, acc, false, false);
  }
  float* Crow = C + (size_t)(tm * 16 + g * 8) * Nd + tn * 16 + col;
#pragma unroll
  for (int i = 0; i < 8; ++i) Crow[(size_t)i * Nd] = acc[i];
}

// ---------------- attention coefficients: a_s/a_d per (node, head) ----------
__global__ void attn_coef(const float* __restrict__ h,
                          const float* __restrict__ att_src,
                          const float* __restrict__ att_dst,
                          float* __restrict__ a_s, float* __restrict__ a_d,
                          int N, int H, int C) {
  int i = blockIdx.x * blockDim.x + threadIdx.x;
  if (i >= N * H) return;
  int n = i / H, hh = i % H;
  const float* hp = h + (size_t)(n * H + hh) * C;
  const float* as = att_src + hh * C;
  const float* ad = att_dst + hh * C;
  float s = 0.f, d = 0.f;
  for (int c = 0; c < C; ++c) {
    float v = hp[c];
    s += v * as[c];
    d += v * ad[c];
  }
  a_s[i] = s;
  a_d[i] = d;
}

// order-preserving float<->uint encoding so atomicMax(u32) == float max,
// with 0 as the identity (0 decodes to -NaN; all reals encode >= 0x007FFFFF)
__device__ __forceinline__ unsigned f2ord(float f) {
  unsigned u = __float_as_uint(f);
  return (u & 0x80000000u) ? ~u : (u | 0x80000000u);
}
__device__ __forceinline__ float ord2f(unsigned u) {
  return (u & 0x80000000u) ? __uint_as_float(u & 0x7FFFFFFFu)
                           : __uint_as_float(~u);
}

__device__ __forceinline__ void edge_endpoints(const long long* ei, long long E,
                                               long long e, int& src, int& dst) {
  if (e < E) {
    src = (int)ei[e];
    dst = (int)ei[E + e];
  } else {
    src = dst = (int)(e - E);  // self-loop
  }
}

// -------- pass A: per-dst running max of leaky_relu(a_s[src]+a_d[dst]) ------
template <int H>
__global__ void edge_max(const long long* __restrict__ ei, long long E, int N,
                         const float* __restrict__ a_s,
                         const float* __restrict__ a_d,
                         unsigned* __restrict__ m_ord) {
  long long i = (long long)blockIdx.x * blockDim.x + threadIdx.x;
  if (i >= (E + N) * H) return;
  long long e = i / H;
  int hh = (int)(i % H);
  int src, dst;
  edge_endpoints(ei, E, e, src, dst);
  float v = a_s[src * H + hh] + a_d[dst * H + hh];
  v = v > 0.f ? v : v * NEG_SLOPE;
  atomicMax(&m_ord[dst * H + hh], f2ord(v));
}

// -------- pass B (fused): p = exp(e - m[dst]); s[dst]+=p; acc[dst]+=p*h[src]
// one wave per edge, lane = channel (C must be 32 = wave32 width)
template <int H, int C>
__global__ void edge_accum(const long long* __restrict__ ei, long long E, int N,
                           const float* __restrict__ a_s,
                           const float* __restrict__ a_d,
                           const unsigned* __restrict__ m_ord,
                           const float* __restrict__ h,
                           float* __restrict__ s_sum,
                           float* __restrict__ acc) {
  static_assert(C == 32, "lane == channel");
  long long gw = ((long long)blockIdx.x * blockDim.x + threadIdx.x) >> 5;
  int lane = threadIdx.x & 31;
  if (gw >= E + N) return;  // wave-uniform
  int src, dst;
  edge_endpoints(ei, E, gw, src, dst);

  float p[H];
#pragma unroll
  for (int hh = 0; hh < H; ++hh) {
    float v = a_s[src * H + hh] + a_d[dst * H + hh];
    v = v > 0.f ? v : v * NEG_SLOPE;
    p[hh] = expf(v - ord2f(m_ord[dst * H + hh]));
    if (lane == 0) atomicAdd(&s_sum[dst * H + hh], p[hh]);
  }
#pragma unroll
  for (int hh = 0; hh < H; ++hh) {
    float msg = p[hh] * h[(size_t)(src * H + hh) * C + lane];
    atomicAdd(&acc[(size_t)(dst * H + hh) * C + lane], msg);
  }
}

// -------- layer-1 finish: out = elu(acc / (s + eps) + bias) -----------------
__global__ void node_finish_elu(const float* __restrict__ acc,
                                const float* __restrict__ s_sum,
                                const float* __restrict__ bias,
                                float* __restrict__ out, int N, int H, int C) {
  int i = blockIdx.x * blockDim.x + threadIdx.x;
  if (i >= N * H * C) return;
  int n = i / (H * C);
  int hc = i % (H * C);
  int hh = hc / C;
  float v = acc[i] / (s_sum[n * H + hh] + 1e-16f) + bias[hc];
  out[i] = v > 0.f ? v : (expf(v) - 1.f);
}

// -------- layer-2 finish fused with predictor head: sigmoid((h2@Wp)+bp) -----
__global__ void node_predict(const float* __restrict__ acc,
                             const float* __restrict__ s_sum,
                             const float* __restrict__ b2,
                             const float* __restrict__ Wp,
                             const float* __restrict__ bp,
                             float* __restrict__ prob, int N, int C) {
  int n = blockIdx.x * blockDim.x + threadIdx.x;
  if (n >= N) return;
  float inv = 1.f / (s_sum[n] + 1e-16f);
  float dot = 0.f;
  for (int c = 0; c < C; ++c)
    dot += (acc[(size_t)n * C + c] * inv + b2[c]) * Wp[c];
  prob[n] = 1.f / (1.f + expf(-(dot + bp[0])));
}

// ---------------------------------------------------------------------------
extern "C" void kernel_launch(void* const* d_in, const int* in_sizes, int n_in,
                              void* d_out, int out_size, void* d_ws,
                              size_t ws_size, hipStream_t stream) {
  const float* x = (const float*)d_in[0];
  const long long* ei = (const long long*)d_in[1];  // int64 edge_index [2,E]
  const float* W1 = (const float*)d_in[2];
  const float* as1w = (const float*)d_in[3];
  const float* ad1w = (const float*)d_in[4];
  const float* b1 = (const float*)d_in[5];
  const float* W2 = (const float*)d_in[6];
  const float* as2w = (const float*)d_in[7];
  const float* ad2w = (const float*)d_in[8];
  const float* b2 = (const float*)d_in[9];
  const float* Wp = (const float*)d_in[10];
  const float* bp = (const float*)d_in[11];
  float* prob = (float*)d_out;

  const int N = in_sizes[0] / 128;        // 50000 nodes
  const long long E = in_sizes[1] / 2;    // 1600000 edges
  const long long ET = E + N;             // with self-loops

  // ---- workspace layout (256B aligned slices) ----
  char* base = (char*)d_ws;
  size_t off = 0;
  auto alloc = [&](size_t bytes) {
    char* p = base + off;
    off += (bytes + 255) & ~(size_t)255;
    return p;
  };
  float* h1 = (float*)alloc((size_t)N * 96 * 4);    // layer-1 features; reused as h2
  float* acc1 = (float*)alloc((size_t)N * 96 * 4);
  float* h2in = (float*)alloc((size_t)N * 96 * 4);  // elu output, input to GEMM2
  float* a_s1 = (float*)alloc((size_t)N * 3 * 4);
  float* a_d1 = (float*)alloc((size_t)N * 3 * 4);
  unsigned* m1 = (unsigned*)alloc((size_t)N * 3 * 4);
  float* s1 = (float*)alloc((size_t)N * 3 * 4);
  float* a_s2 = (float*)alloc((size_t)N * 4);
  float* a_d2 = (float*)alloc((size_t)N * 4);
  unsigned* m2 = (unsigned*)alloc((size_t)N * 4);
  float* s2 = (float*)alloc((size_t)N * 4);
  float* acc2 = (float*)alloc((size_t)N * 32 * 4);
  float* h2 = h1;  // safe reuse: h1 dead after layer-1 edge_accum

  // ---- zero the reduction buffers (0 is identity for both encodings) ----
  hipMemsetAsync(m1, 0, (size_t)N * 3 * 4, stream);
  hipMemsetAsync(s1, 0, (size_t)N * 3 * 4, stream);
  hipMemsetAsync(acc1, 0, (size_t)N * 96 * 4, stream);
  hipMemsetAsync(m2, 0, (size_t)N * 4, stream);
  hipMemsetAsync(s2, 0, (size_t)N * 4, stream);
  hipMemsetAsync(acc2, 0, (size_t)N * 32 * 4, stream);

  const int BLK = 256;

  // ================= layer 1 (H=3, C=32) =================
  {
    long long warps = (long long)(N / 16) * (96 / 16);
    gemm_f32_wmma<96, 128><<<cdiv(warps * 32, BLK), BLK, 0, stream>>>(x, W1, h1, N);
  }
  attn_coef<<<cdiv((long long)N * 3, BLK), BLK, 0, stream>>>(h1, as1w, ad1w,
                                                             a_s1, a_d1, N, 3, 32);
  edge_max<3><<<cdiv(ET * 3, BLK), BLK, 0, stream>>>(ei, E, N, a_s1, a_d1, m1);
  edge_accum<3, 32><<<cdiv(ET * 32, BLK), BLK, 0, stream>>>(ei, E, N, a_s1, a_d1,
                                                            m1, h1, s1, acc1);
  node_finish_elu<<<cdiv((long long)N * 96, BLK), BLK, 0, stream>>>(acc1, s1, b1,
                                                                    h2in, N, 3, 32);

  // ================= layer 2 (H=1, C=32) =================
  {
    long long warps = (long long)(N / 16) * (32 / 16);
    gemm_f32_wmma<32, 96><<<cdiv(warps * 32, BLK), BLK, 0, stream>>>(h2in, W2, h2, N);
  }
  attn_coef<<<cdiv((long long)N, BLK), BLK, 0, stream>>>(h2, as2w, ad2w,
                                                         a_s2, a_d2, N, 1, 32);
  edge_max<1><<<cdiv(ET, BLK), BLK, 0, stream>>>(ei, E, N, a_s2, a_d2, m2);
  edge_accum<1, 32><<<cdiv(ET * 32, BLK), BLK, 0, stream>>>(ei, E, N, a_s2, a_d2,
                                                            m2, h2, s2, acc2);

  // ================= predictor head =================
  node_predict<<<cdiv((long long)N, BLK), BLK, 0, stream>>>(acc2, s2, b2, Wp, bp,
                                                            prob, N, 32);
  (void)n_in; (void)out_size; (void)ws_size;
}